// YawSearcher_68719477348
// MI455X (gfx1250) — compile-verified
//
#include <hip/hip_runtime.h>
#include <hip/hip_bf16.h>

typedef __attribute__((ext_vector_type(2))) float v2f;
typedef __attribute__((ext_vector_type(8))) float v8f;

#define N_BINS 1024
#define DIM 512
#define N_STEPS 240
#define KSLOTS 128   // shift slots k in [-64, 63], slot = k + 64

// ---------------------------------------------------------------------------
// Kernel 1: banded Gram matrix C[n][k+64] = D_p[n] . D_q[(n-k) mod 1024]
// via fp32 WMMA (v_wmma_f32_16x16x4_f32). One wave per 16x16 output tile.
// 64 row tiles x 9 column tiles (covers circular band k in [-61,60]).
// ---------------------------------------------------------------------------
__global__ __launch_bounds__(256) void band_gemm_kernel(
    const float* __restrict__ Dq, const float* __restrict__ Dp,
    float* __restrict__ C) {
  const int wave = threadIdx.x >> 5;
  const int lane = threadIdx.x & 31;
  const int tile = blockIdx.x * (blockDim.x >> 5) + wave;
  if (tile >= 64 * 9) return;  // uniform per wave

  const int i  = tile / 9;             // row tile of D_p
  const int jj = tile - i * 9;         // 0..8 band column offset
  const int j  = (i - 4 + jj + 64) & 63;  // column tile of D_q (circular)
  const int row0 = i << 4;
  const int col0 = j << 4;

  const int half = lane >> 4;          // K sub-block: lanes 0-15 -> K0..1, 16-31 -> K2..3
  const int lm   = lane & 15;

  // A = D_p rows (16xK), B = D_q rows as K x 16 (i.e., D_q^T tile)
  const float* aptr = Dp + (size_t)(row0 + lm) * DIM + 2 * half;
  const float* bptr = Dq + (size_t)(col0 + lm) * DIM + 2 * half;

  __builtin_prefetch(aptr, 0, 0);      // global_prefetch_b8 (speculative, L2)
  __builtin_prefetch(bptr, 0, 0);

  v8f acc = {};
#pragma unroll 4
  for (int kk = 0; kk < DIM; kk += 4) {
    v2f av = *(const v2f*)(aptr + kk);
    v2f bv = *(const v2f*)(bptr + kk);
    // D = A(16x4,f32) x B(4x16,f32) + C(16x16,f32)
    acc = __builtin_amdgcn_wmma_f32_16x16x4_f32(
        /*neg_a=*/false, av, /*neg_b=*/false, bv,
        /*c_mod=*/(short)0, acc, /*reuse_a=*/false, /*reuse_b=*/false);
  }

  // C/D layout: VGPR r, lanes 0-15 -> (M=r, N=lane); lanes 16-31 -> (M=r+8, N=lane-16)
  const int m = col0 + lm;             // D_q row index
#pragma unroll
  for (int r = 0; r < 8; ++r) {
    const int n = row0 + r + (half << 3);                    // D_p row index
    const int ksh = (int)((unsigned)(n - m + 1536) & 1023u) - 512;  // centered shift
    if (ksh >= -64 && ksh < 64) {
      C[((size_t)n << 7) + (ksh + 64)] = acc[r];
    }
  }
}

// ---------------------------------------------------------------------------
// Kernel 2: per-row self terms: ||q_j||^2, ||p_n||^2, adj[j] = q_j . q_{j+1}
// ---------------------------------------------------------------------------
__global__ __launch_bounds__(128) void norms_kernel(
    const float* __restrict__ Dq, const float* __restrict__ Dp,
    float* __restrict__ qn, float* __restrict__ pn, float* __restrict__ adj) {
  const int row = blockIdx.x;
  const int nxt = (row + 1) & (N_BINS - 1);
  const float* q0 = Dq + (size_t)row * DIM;
  const float* q1 = Dq + (size_t)nxt * DIM;
  const float* p  = Dp + (size_t)row * DIM;

  float sq = 0.f, sp = 0.f, sa = 0.f;
  for (int d = threadIdx.x; d < DIM; d += blockDim.x) {
    const float qa = q0[d], qb = q1[d], pv = p[d];
    sq = fmaf(qa, qa, sq);
    sp = fmaf(pv, pv, sp);
    sa = fmaf(qa, qb, sa);
  }
  __shared__ float r0[128], r1[128], r2[128];
  const int t = threadIdx.x;
  r0[t] = sq; r1[t] = sp; r2[t] = sa;
  __syncthreads();
  for (int off = 64; off > 0; off >>= 1) {
    if (t < off) { r0[t] += r0[t + off]; r1[t] += r1[t + off]; r2[t] += r2[t + off]; }
    __syncthreads();
  }
  if (t == 0) { qn[row] = r0[0]; pn[row] = r1[0]; adj[row] = r2[0]; }
}

// ---------------------------------------------------------------------------
// Kernel 3: per-step distance via closed form; one block per step.
// d2(n) = ||p_n||^2 - 2*((1-a)C[n][k0] + a*C[n][k1])
//         + (1-a)^2 ||q_m0||^2 + a^2 ||q_m1||^2 + 2a(1-a) q_m0.q_m1
// ---------------------------------------------------------------------------
__global__ __launch_bounds__(256) void step_dist_kernel(
    const float* __restrict__ C, const float* __restrict__ qn,
    const float* __restrict__ pn, const float* __restrict__ adj,
    float* __restrict__ dists) {
  const int s = blockIdx.x;
  float step; int sign;
  if (s < 120) { step = 0.5f * (float)s;                sign = 1;  }
  else         { step = -(1.0f + 0.5f * (float)(s-120)); sign = -1; }
  const float a     = fabsf(step);
  const int   k     = (int)a;
  const float alpha = a - (float)k;
  const int   k0    = sign * k;
  const int   slot0 = k0 + 64;
  const int   slot1 = k0 + sign + 64;
  const float w0 = 1.0f - alpha, w1 = alpha;
  const float w00 = w0 * w0, w11 = w1 * w1, w01 = 2.0f * w0 * w1;

  float partial = 0.0f;
  for (int n = threadIdx.x; n < N_BINS; n += blockDim.x) {
    const int m0 = (n - k0 + N_BINS) & (N_BINS - 1);
    const int m1 = (m0 - sign + N_BINS) & (N_BINS - 1);
    const int ai = (sign > 0) ? ((m0 + N_BINS - 1) & (N_BINS - 1)) : m0;
    const float cross = w0 * C[((size_t)n << 7) + slot0] +
                        w1 * C[((size_t)n << 7) + slot1];
    float d2 = pn[n] - 2.0f * cross + w00 * qn[m0] + w11 * qn[m1] + w01 * adj[ai];
    partial += sqrtf(fmaxf(d2, 0.0f));
  }
  __shared__ float red[256];
  const int t = threadIdx.x;
  red[t] = partial;
  __syncthreads();
  for (int off = 128; off > 0; off >>= 1) {
    if (t < off) red[t] += red[t + off];
    __syncthreads();
  }
  if (t == 0) dists[s] = red[0] * (1.0f / (float)N_BINS);
}

// ---------------------------------------------------------------------------
// Kernel 4: argmin over 240 steps (first-index tie-break like jnp.argmin)
// ---------------------------------------------------------------------------
__global__ __launch_bounds__(256) void argmin_kernel(
    const float* __restrict__ dists, float* __restrict__ out) {
  __shared__ float sv[256];
  __shared__ int   si[256];
  const int t = threadIdx.x;
  sv[t] = (t < N_STEPS) ? dists[t] : 3.4e38f;
  si[t] = t;
  __syncthreads();
  for (int off = 128; off > 0; off >>= 1) {
    if (t < off) {
      const float vb = sv[t + off];
      const int   ib = si[t + off];
      if (vb < sv[t] || (vb == sv[t] && ib < si[t])) { sv[t] = vb; si[t] = ib; }
    }
    __syncthreads();
  }
  if (t == 0) {
    const int s = si[0];
    out[0] = (s < 120) ? 0.5f * (float)s : -(1.0f + 0.5f * (float)(s - 120));
    out[1] = sv[0];
  }
}

// ---------------------------------------------------------------------------
extern "C" void kernel_launch(void* const* d_in, const int* in_sizes, int n_in,
                              void* d_out, int out_size, void* d_ws, size_t ws_size,
                              hipStream_t stream) {
  const float* Dq = (const float*)d_in[0];  // D_q [1024,512] fp32
  const float* Dp = (const float*)d_in[1];  // D_p [1024,512] fp32

  float* ws    = (float*)d_ws;
  float* C     = ws;                        // 1024*128 = 131072 floats (512 KB)
  float* qn    = C + (size_t)N_BINS * KSLOTS;
  float* pn    = qn + N_BINS;
  float* adj   = pn + N_BINS;
  float* dists = adj + N_BINS;              // 240 floats

  // 576 WMMA tiles, 8 waves (tiles) per 256-thread block -> 72 blocks
  band_gemm_kernel<<<72, 256, 0, stream>>>(Dq, Dp, C);
  norms_kernel<<<N_BINS, 128, 0, stream>>>(Dq, Dp, qn, pn, adj);
  step_dist_kernel<<<N_STEPS, 256, 0, stream>>>(C, qn, pn, adj, dists);
  argmin_kernel<<<1, 256, 0, stream>>>(dists, (float*)d_out);
}